// RegimeFactorModel_63110249447761
// MI455X (gfx1250) — compile-verified
//
#include <hip/hip_runtime.h>

typedef __attribute__((ext_vector_type(2))) float v2f;
typedef __attribute__((ext_vector_type(8))) float v8f;

#define NN 2048
#define FD 64
#define KK 8
#define BB 32

// Packed 2xf32 FMA (VOP3P V_PK_FMA_F32): d = a*b + c per 32-bit half.
// All operands in VGPRs (gfx1250 assembler rejects SGPR-pair packed sources).
static __device__ __forceinline__ v2f pk_fma(v2f a, v2f b, v2f c) {
    v2f d;
    asm("v_pk_fma_f32 %0, %1, %2, %3" : "=v"(d) : "v"(a), "v"(b), "v"(c));
    return d;
}

// Fused + symmetric: out[b] = sum_k p[b,k] * (L_k L_k^T + diag(S_k^2)) is
// symmetric, so only tiles with col0 >= row0 are computed; off-diagonal tiles
// are stored twice (direct + mirrored). The mirrored write is the well-
// coalesced one: each lane owns 8 consecutive columns -> NT b64 stores taken
// directly from the pk_fma result pairs (no repacking movs).
__global__ __launch_bounds__(256) void regime_factor_fused(
    const float* __restrict__ P,   // (B, K)   regime_probs
    const float* __restrict__ L,   // (K, N, F) factor_loadings
    const float* __restrict__ S,   // (K, N)   specific_risk
    float* __restrict__ out)       // (B, N, N)
{
    const int lane  = threadIdx.x & 31;
    const int wave  = threadIdx.x >> 5;
    const int row0  = blockIdx.y * 16;                 // tile row base (M)
    const int col0  = blockIdx.x * 128 + wave * 16;    // tile col base (N)

    // Symmetry: lower-triangle tiles are produced as mirrors of upper ones.
    // Condition is wave-uniform, so surviving waves have EXEC all-ones.
    if (col0 < row0) return;

    // WMMA f32 16x16x4 operand layout (wave32):
    //   A (16x4):  lane l -> row M = l&15 ; VGPR0/1 hold K = (l>>4)*2 + {0,1}
    //   B (4x16):  lane l -> col N = l&15 ; VGPR0/1 hold K = (l>>4)*2 + {0,1}
    const int mn    = lane & 15;
    const int khalf = lane >> 4;          // 0 or 1
    const int koff  = khalf * 2;

    // cov tiles held as float2 pairs: cov2[k][j] = {C[2j], C[2j+1]} in C layout.
    v2f cov2[KK][4];

#pragma unroll
    for (int k = 0; k < KK; ++k) {
        const float* La = L + ((size_t)k * NN + (size_t)(row0 + mn)) * FD + koff;
        const float* Lb = L + ((size_t)k * NN + (size_t)(col0 + mn)) * FD + koff;
        v8f acc = {};
#pragma unroll
        for (int f = 0; f < FD; f += 4) {
            v2f a = *(const v2f*)(La + f);
            v2f b = *(const v2f*)(Lb + f);
            // (neg_a, A, neg_b, B, c_mod, C, reuse_a, reuse_b)
            acc = __builtin_amdgcn_wmma_f32_16x16x4_f32(
                false, a, false, b, (short)0, acc, false, false);
        }
        // Diagonal tiles: add specific_risk^2 where global row == global col.
        if (row0 == col0) {
#pragma unroll
            for (int v = 0; v < 8; ++v) {
                const int m = v + (khalf << 3);   // M index of this VGPR in C layout
                if (m == mn) {
                    float s = S[(size_t)k * NN + (size_t)(row0 + m)];
                    acc[v] += s * s;
                }
            }
        }
#pragma unroll
        for (int j = 0; j < 4; ++j) {
            v2f c; c.x = acc[2 * j]; c.y = acc[2 * j + 1];
            cov2[k][j] = c;
        }
    }

    // Combine: out[b] tile = sum_k p[b,k] * cov[k] via v_pk_fma_f32.
    // Unroll-by-4 keeps code size moderate and lets the 32 consecutive P
    // scalar loads per iteration vectorize into wide s_loads.
    // C/D layout: VGPR v, lanes 0-15 -> M=v, lanes 16-31 -> M=v+8; N = lane&15.
    const size_t colidx  = (size_t)(col0 + mn);
    const int    rbase   = row0 + (khalf << 3);    // first of 8 consecutive rows
    const bool   offdiag = (row0 != col0);

#pragma unroll 4
    for (int b = 0; b < BB; ++b) {
        v2f o[4] = {{0.f, 0.f}, {0.f, 0.f}, {0.f, 0.f}, {0.f, 0.f}};
#pragma unroll
        for (int k = 0; k < KK; ++k) {
            const float pw = P[b * KK + k];   // wave-uniform -> scalar load
            v2f pw2; pw2.x = pw; pw2.y = pw;  // broadcast pair (built once per k)
#pragma unroll
            for (int j = 0; j < 4; ++j) o[j] = pk_fma(pw2, cov2[k][j], o[j]);
        }
        float* ob = out + (size_t)b * NN * NN;

        // Direct tile at (row0, col0): lane owns one col across 8 rows.
#pragma unroll
        for (int j = 0; j < 4; ++j) {
            const int r0 = rbase + 2 * j;
            __builtin_nontemporal_store(o[j].x, ob + (size_t)r0 * NN + colidx);
            __builtin_nontemporal_store(o[j].y, ob + (size_t)(r0 + 1) * NN + colidx);
        }

        // Mirrored tile at (col0, row0): lane owns 8 consecutive cols in one
        // row -> four NT b64 stores straight from the pk_fma result pairs.
        if (offdiag) {
            float* tb = ob + colidx * NN + (size_t)rbase;
#pragma unroll
            for (int j = 0; j < 4; ++j)
                __builtin_nontemporal_store(o[j], (v2f*)(tb + 2 * j));
        }
    }
}

extern "C" void kernel_launch(void* const* d_in, const int* in_sizes, int n_in,
                              void* d_out, int out_size, void* d_ws, size_t ws_size,
                              hipStream_t stream) {
    const float* P = (const float*)d_in[0];   // (32, 8)
    const float* L = (const float*)d_in[1];   // (8, 2048, 64)
    const float* S = (const float*)d_in[2];   // (8, 2048)
    float* out     = (float*)d_out;           // (32, 2048, 2048)
    (void)in_sizes; (void)n_in; (void)d_ws; (void)ws_size; (void)out_size;

    dim3 grid(NN / 128, NN / 16);   // (16, 128): x = col blocks of 128, y = row tiles of 16
    dim3 block(256);                // 8 waves, each owns one 16x16 tile
    hipLaunchKernelGGL(regime_factor_fused, grid, block, 0, stream, P, L, S, out);
}